// GCNWMMSE_81750407512802
// MI455X (gfx1250) — compile-verified
//
#include <hip/hip_runtime.h>
#include <hip/hip_bf16.h>

// ---------------------------------------------------------------------------
// GCN-WMMSE iteration for MI455X (gfx1250, wave32).
// B=32 batches, C=8 BSs, K=32 users, M=128 antennas, N=4 streams, complex fp32.
// Heavy 128x128x128 complex GEMMs use V_WMMA_F32_16X16X4_F32 with operand
// tiles staged into LDS via CDNA5 async global->LDS copies (ASYNCcnt path).
// 128x128 Hermitian eigendecomposition: parallel cyclic Jacobi fully in LDS
// (264KB per workgroup; CDNA5 WGP has 320KB LDS).
// ---------------------------------------------------------------------------

#define BATCH 32
#define NBSS  8
#define NUSR  32
#define MDIM  128
#define NDIM  4
#define QDIM  128              // NUSR * NDIM
#define BC    (BATCH*NBSS)     // 256
#define BK    (BATCH*NUSR)     // 1024
#define MAT   (MDIM*MDIM)      // 16384
#define NSWEEP 8
#define EP    129              // eig LDS pitch (odd -> bank-conflict free)
#define EIG_LDS_FLOATS (4*MDIM*EP)   // Ar,Ai,Er,Ei
#define EIG_LDS_BYTES  (EIG_LDS_FLOATS*4)
#define AP    132              // GEMM A-strip LDS pitch: 16B-aligned rows,
                               // 4-bank rotation per row -> conflict-free

typedef float v2f __attribute__((ext_vector_type(2)));
typedef float v8f __attribute__((ext_vector_type(8)));

static __device__ __forceinline__ v8f wmma_f32x4(v2f a, v2f b, v8f c) {
  // D(16x16,f32) += A(16x4,f32) * B(4x16,f32)
  return __builtin_amdgcn_wmma_f32_16x16x4_f32(false, a, false, b, (short)0, c,
                                               false, false);
}

// CDNA5 async global->LDS copy: each active lane moves 16B from its global
// address to its LDS byte offset. Tracked by ASYNCcnt.
static __device__ __forceinline__ void async_copy16(unsigned int lds_off,
                                                    const float* gptr) {
  asm volatile("global_load_async_to_lds_b128 %0, %1, off"
               :
               : "v"(lds_off), "v"((unsigned long long)(uintptr_t)gptr)
               : "memory");
}

static __device__ __forceinline__ void wait_async0() {
  asm volatile("s_wait_asynccnt 0" ::: "memory");
}

// LDS byte offset of a __shared__ object (flat LDS addr low 32 bits == offset)
static __device__ __forceinline__ unsigned int lds_off_of(const void* p) {
  return (unsigned int)(uintptr_t)p;
}

// ---------------------------------------------------------------------------
// K0: transpose H[c] (viewed as QDIM x MDIM) -> HT[c] (MDIM x QDIM)
// ---------------------------------------------------------------------------
__global__ void k_transposeH(const float* __restrict__ Hre,
                             const float* __restrict__ Him,
                             float* __restrict__ HTre,
                             float* __restrict__ HTim) {
  int c = blockIdx.x >> 7;
  int m = blockIdx.x & 127;
  int q = threadIdx.x;
  HTre[c*MAT + m*MDIM + q] = Hre[c*MAT + q*MDIM + m];
  HTim[c*MAT + m*MDIM + q] = Him[c*MAT + q*MDIM + m];
}

// ---------------------------------------------------------------------------
// K1: downlink covariance per (b,i) and direct-channel HdV.
// cov[b,i] = noise*I + sum_{j,p} z z^H,  z = (H[assign[j],i] V[b,j])[:,p]
// ---------------------------------------------------------------------------
__global__ void k_cov(const float* __restrict__ vre, const float* __restrict__ vim,
                      const float* __restrict__ Hre, const float* __restrict__ Him,
                      const float* __restrict__ noise, const int* __restrict__ asg,
                      float* __restrict__ covr, float* __restrict__ covi,
                      float* __restrict__ hdvr, float* __restrict__ hdvi) {
  int b = blockIdx.x >> 5;
  int i = blockIdx.x & 31;
  int t = threadIdx.x;          // 128 threads: t = (j, p)
  int j = t >> 2;
  int p = t & 3;
  __shared__ float zr[4*128], zi[4*128];

  int cj = asg[j];
  const float* hbR = Hre + ((cj*NUSR + i)*NDIM)*MDIM;
  const float* hbI = Him + ((cj*NUSR + i)*NDIM)*MDIM;
  const float* vbR = vre + ((size_t)(b*NUSR + j)*MDIM)*NDIM + p;
  const float* vbI = vim + ((size_t)(b*NUSR + j)*MDIM)*NDIM + p;

  float ar0=0,ar1=0,ar2=0,ar3=0, ai0=0,ai1=0,ai2=0,ai3=0;
  for (int m = 0; m < MDIM; ++m) {
    float vr = vbR[m*4], vi_ = vbI[m*4];
    float hr, hi;
    hr = hbR[0*MDIM+m]; hi = hbI[0*MDIM+m]; ar0 += hr*vr - hi*vi_; ai0 += hr*vi_ + hi*vr;
    hr = hbR[1*MDIM+m]; hi = hbI[1*MDIM+m]; ar1 += hr*vr - hi*vi_; ai1 += hr*vi_ + hi*vr;
    hr = hbR[2*MDIM+m]; hi = hbI[2*MDIM+m]; ar2 += hr*vr - hi*vi_; ai2 += hr*vi_ + hi*vr;
    hr = hbR[3*MDIM+m]; hi = hbI[3*MDIM+m]; ar3 += hr*vr - hi*vi_; ai3 += hr*vi_ + hi*vr;
  }
  zr[0*128+t]=ar0; zr[1*128+t]=ar1; zr[2*128+t]=ar2; zr[3*128+t]=ar3;
  zi[0*128+t]=ai0; zi[1*128+t]=ai1; zi[2*128+t]=ai2; zi[3*128+t]=ai3;
  if (j == i) {   // direct-channel product HdV[b,i][n][p]
    int base = (b*NUSR + i)*16;
    hdvr[base + 0*4 + p] = ar0; hdvi[base + 0*4 + p] = ai0;
    hdvr[base + 1*4 + p] = ar1; hdvi[base + 1*4 + p] = ai1;
    hdvr[base + 2*4 + p] = ar2; hdvi[base + 2*4 + p] = ai2;
    hdvr[base + 3*4 + p] = ar3; hdvi[base + 3*4 + p] = ai3;
  }
  __syncthreads();
  if (t < 16) {
    int n = t >> 2, q = t & 3;
    float sr = 0.f, si = 0.f;
    for (int col = 0; col < 128; ++col) {
      float xr = zr[n*128+col], xi = zi[n*128+col];
      float yr = zr[q*128+col], yi = zi[q*128+col];
      sr += xr*yr + xi*yi;        // x * conj(y)
      si += xi*yr - xr*yi;
    }
    if (n == q) sr += noise[b*NUSR + i];
    covr[(b*NUSR+i)*16 + n*4 + q] = sr;
    covi[(b*NUSR+i)*16 + n*4 + q] = si;
  }
}

// ---------------------------------------------------------------------------
// 4x4 complex helpers (per-thread, scalar)
// ---------------------------------------------------------------------------
static __device__ void cherm4(float xr[4][4], float xi[4][4]) {
  float tr[4][4], ti[4][4];
  for (int a = 0; a < 4; ++a) for (int b = 0; b < 4; ++b) { tr[a][b]=xr[a][b]; ti[a][b]=xi[a][b]; }
  for (int a = 0; a < 4; ++a) for (int b = 0; b < 4; ++b) {
    xr[a][b] = 0.5f*(tr[a][b] + tr[b][a]);
    xi[a][b] = 0.5f*(ti[a][b] - ti[b][a]);
  }
}

static __device__ void cinv4(float mr[4][4], float mi[4][4],
                             float rr[4][4], float ri[4][4]) {
  for (int a = 0; a < 4; ++a) for (int b = 0; b < 4; ++b) {
    rr[a][b] = (a == b) ? 1.f : 0.f; ri[a][b] = 0.f;
  }
  for (int col = 0; col < 4; ++col) {
    int piv = col;
    float best = mr[col][col]*mr[col][col] + mi[col][col]*mi[col][col];
    for (int r = col+1; r < 4; ++r) {
      float v = mr[r][col]*mr[r][col] + mi[r][col]*mi[r][col];
      if (v > best) { best = v; piv = r; }
    }
    if (piv != col) {
      for (int c2 = 0; c2 < 4; ++c2) {
        float t;
        t = mr[col][c2]; mr[col][c2] = mr[piv][c2]; mr[piv][c2] = t;
        t = mi[col][c2]; mi[col][c2] = mi[piv][c2]; mi[piv][c2] = t;
        t = rr[col][c2]; rr[col][c2] = rr[piv][c2]; rr[piv][c2] = t;
        t = ri[col][c2]; ri[col][c2] = ri[piv][c2]; ri[piv][c2] = t;
      }
    }
    float dr = mr[col][col], di = mi[col][col];
    float d = dr*dr + di*di + 1e-30f;
    float pr = dr/d, pi_ = -di/d;         // 1/pivot
    for (int c2 = 0; c2 < 4; ++c2) {
      float xr = mr[col][c2], xi = mi[col][c2];
      mr[col][c2] = xr*pr - xi*pi_;  mi[col][c2] = xr*pi_ + xi*pr;
      xr = rr[col][c2]; xi = ri[col][c2];
      rr[col][c2] = xr*pr - xi*pi_;  ri[col][c2] = xr*pi_ + xi*pr;
    }
    for (int r = 0; r < 4; ++r) {
      if (r == col) continue;
      float fr = mr[r][col], fi = mi[r][col];
      for (int c2 = 0; c2 < 4; ++c2) {
        mr[r][c2] -= fr*mr[col][c2] - fi*mi[col][c2];
        mi[r][c2] -= fr*mi[col][c2] + fi*mr[col][c2];
        rr[r][c2] -= fr*rr[col][c2] - fi*ri[col][c2];
        ri[r][c2] -= fr*ri[col][c2] + fi*rr[col][c2];
      }
    }
  }
}

// ---------------------------------------------------------------------------
// K2: per (b,k): icov -> u -> w -> uw, uwu
// ---------------------------------------------------------------------------
__global__ void k_small(const float* __restrict__ covr, const float* __restrict__ covi,
                        const float* __restrict__ hdvr, const float* __restrict__ hdvi,
                        float* __restrict__ uwr, float* __restrict__ uwi,
                        float* __restrict__ uwur, float* __restrict__ uwui) {
  int bk = blockIdx.x*blockDim.x + threadIdx.x;
  if (bk >= BK) return;
  float ar[4][4], ai[4][4], icr[4][4], ici[4][4];
  float hr[4][4], hi[4][4];
  for (int a = 0; a < 4; ++a) for (int b = 0; b < 4; ++b) {
    ar[a][b] = covr[bk*16 + a*4 + b]; ai[a][b] = covi[bk*16 + a*4 + b];
    hr[a][b] = hdvr[bk*16 + a*4 + b]; hi[a][b] = hdvi[bk*16 + a*4 + b];
  }
  cinv4(ar, ai, icr, ici);
  cherm4(icr, ici);
  // u = icov @ hdv
  float ur[4][4], ui[4][4];
  for (int n = 0; n < 4; ++n) for (int p = 0; p < 4; ++p) {
    float sr = 0.f, si = 0.f;
    for (int q = 0; q < 4; ++q) {
      sr += icr[n][q]*hr[q][p] - ici[n][q]*hi[q][p];
      si += icr[n][q]*hi[q][p] + ici[n][q]*hr[q][p];
    }
    ur[n][p] = sr; ui[n][p] = si;
  }
  // winv = I - u^H hdv
  float wr[4][4], wi[4][4];
  for (int n = 0; n < 4; ++n) for (int p = 0; p < 4; ++p) {
    float sr = 0.f, si = 0.f;
    for (int q = 0; q < 4; ++q) {
      sr += ur[q][n]*hr[q][p] + ui[q][n]*hi[q][p];
      si += ur[q][n]*hi[q][p] - ui[q][n]*hr[q][p];
    }
    wr[n][p] = ((n==p)?1.f:0.f) - sr;
    wi[n][p] = -si;
  }
  cherm4(wr, wi);
  float Wr[4][4], Wi[4][4];
  cinv4(wr, wi, Wr, Wi);
  cherm4(Wr, Wi);
  // uw = u @ W
  float pwr[4][4], pwi[4][4];
  for (int n = 0; n < 4; ++n) for (int p = 0; p < 4; ++p) {
    float sr = 0.f, si = 0.f;
    for (int q = 0; q < 4; ++q) {
      sr += ur[n][q]*Wr[q][p] - ui[n][q]*Wi[q][p];
      si += ur[n][q]*Wi[q][p] + ui[n][q]*Wr[q][p];
    }
    pwr[n][p] = sr; pwi[n][p] = si;
  }
  // uwu = uw @ u^H
  float tr[4][4], ti[4][4];
  for (int n = 0; n < 4; ++n) for (int r = 0; r < 4; ++r) {
    float sr = 0.f, si = 0.f;
    for (int q = 0; q < 4; ++q) {
      sr += pwr[n][q]*ur[r][q] + pwi[n][q]*ui[r][q];
      si += pwi[n][q]*ur[r][q] - pwr[n][q]*ui[r][q];
    }
    tr[n][r] = sr; ti[n][r] = si;
  }
  cherm4(tr, ti);
  for (int a = 0; a < 4; ++a) for (int b = 0; b < 4; ++b) {
    uwr [bk*16 + a*4 + b] = pwr[a][b];  uwi [bk*16 + a*4 + b] = pwi[a][b];
    uwur[bk*16 + a*4 + b] = tr[a][b];   uwui[bk*16 + a*4 + b] = ti[a][b];
  }
}

// ---------------------------------------------------------------------------
// K3: vt = rw * Hd^H (u w); stored stacked & conjugated: S[b][q=(k,p)][m] = conj(vt)
// ---------------------------------------------------------------------------
__global__ void k_vt(const float* __restrict__ Hre, const float* __restrict__ Him,
                     const float* __restrict__ uwr, const float* __restrict__ uwi,
                     const float* __restrict__ rweights, const int* __restrict__ asg,
                     float* __restrict__ Sre, float* __restrict__ Sim) {
  int bk = blockIdx.x;
  int b = bk >> 5, k = bk & 31;
  int m = threadIdx.x;
  int c0 = asg[k];
  float rwv = rweights[bk];
  const float* hbR = Hre + ((c0*NUSR + k)*NDIM)*MDIM + m;
  const float* hbI = Him + ((c0*NUSR + k)*NDIM)*MDIM + m;
  float hr[4], hi[4];
  for (int n = 0; n < 4; ++n) { hr[n] = hbR[n*MDIM]; hi[n] = hbI[n*MDIM]; }
  const float* pr = uwr + bk*16;
  const float* pi_ = uwi + bk*16;
  for (int p = 0; p < 4; ++p) {
    float vr = 0.f, vi = 0.f;
    for (int n = 0; n < 4; ++n) {
      float arr = hr[n], aii = -hi[n];            // conj(Hd[n][m])
      float br = pr[n*4+p], bi = pi_[n*4+p];
      vr += arr*br - aii*bi;
      vi += arr*bi + aii*br;
    }
    vr *= rwv; vi *= rwv;
    Sre[(size_t)b*MAT + (k*4+p)*MDIM + m] = vr;    // conj(vt): re
    Sim[(size_t)b*MAT + (k*4+p)*MDIM + m] = -vi;   // conj(vt): im
  }
}

// ---------------------------------------------------------------------------
// K4 (WMMA + async LDS staging): ul[b,c] = Hs^H * G, Hs = H[c] as 128x128 [q][m].
// A-strip (HT rows m-block, contiguous 8KB/plane) staged via async copies;
// G rows built on the fly = rw * uwu[b,kk] @ H[c,kk]. One wave per 16x16 tile.
// ---------------------------------------------------------------------------
__global__ void k_ul(const float* __restrict__ HTre, const float* __restrict__ HTim,
                     const float* __restrict__ Hre,  const float* __restrict__ Him,
                     const float* __restrict__ uwur, const float* __restrict__ uwui,
                     const float* __restrict__ rweights,
                     float* __restrict__ ULre, float* __restrict__ ULim) {
  int t = blockIdx.x;
  int tile = t & 63;
  int bc = t >> 6;
  int b = bc >> 3, c = bc & 7;
  int trow = tile >> 3, tcol = tile & 7;
  int l  = threadIdx.x;
  int lr = l & 15;
  int kb = (l >> 4) << 1;                 // K-sub-index {0,2}
  int col = tcol*16 + lr;                 // B (output col)
  const float* HTreC = HTre + c*MAT;
  const float* HTimC = HTim + c*MAT;

  __shared__ float sAre[16*AP], sAim[16*AP];
  {
    // stage the 16x128 A-strip (re+im): one async b128 per lane per row
    const float* gre = HTreC + (trow*16)*MDIM;
    const float* gim = HTimC + (trow*16)*MDIM;
    unsigned int oRe = lds_off_of(sAre);
    unsigned int oIm = lds_off_of(sAim);
    for (int r = 0; r < 16; ++r) {
      async_copy16(oRe + (unsigned)(r*AP + l*4)*4u, gre + r*MDIM + l*4);
      async_copy16(oIm + (unsigned)(r*AP + l*4)*4u, gim + r*MDIM + l*4);
    }
  }
  wait_async0();

  v8f accR = {}, accIp = {}, accIm = {};
  for (int kk = 0; kk < NUSR; ++kk) {
    int q0 = kk*4;
    v2f a_re = *(const v2f*)(sAre + lr*AP + q0 + kb);
    v2f a_im = *(const v2f*)(sAim + lr*AP + q0 + kb);

    const float* ur = uwur + (b*NUSR + kk)*16;    // uniform -> SGPR
    const float* ui = uwui + (b*NUSR + kk)*16;
    float rw = rweights[b*NUSR + kk];

    const float* hrb = Hre + ((c*NUSR + kk)*NDIM)*MDIM + col;
    const float* hib = Him + ((c*NUSR + kk)*NDIM)*MDIM + col;
    float hr0 = hrb[0], hr1 = hrb[MDIM], hr2 = hrb[2*MDIM], hr3 = hrb[3*MDIM];
    float hi0 = hib[0], hi1 = hib[MDIM], hi2 = hib[2*MDIM], hi3 = hib[3*MDIM];
    // prefetch next user's channel rows (speculative; dropped if unmapped)
    __builtin_prefetch(hrb + NDIM*MDIM, 0, 0);
    __builtin_prefetch(hib + NDIM*MDIM, 0, 0);

    int r0 = kb*4, r1 = (kb+1)*4;
    float xr0 = ur[r0+0]*hr0 - ui[r0+0]*hi0 + ur[r0+1]*hr1 - ui[r0+1]*hi1
              + ur[r0+2]*hr2 - ui[r0+2]*hi2 + ur[r0+3]*hr3 - ui[r0+3]*hi3;
    float xi0 = ur[r0+0]*hi0 + ui[r0+0]*hr0 + ur[r0+1]*hi1 + ui[r0+1]*hr1
              + ur[r0+2]*hi2 + ui[r0+2]*hr2 + ur[r0+3]*hi3 + ui[r0+3]*hr3;
    float xr1 = ur[r1+0]*hr0 - ui[r1+0]*hi0 + ur[r1+1]*hr1 - ui[r1+1]*hi1
              + ur[r1+2]*hr2 - ui[r1+2]*hi2 + ur[r1+3]*hr3 - ui[r1+3]*hi3;
    float xi1 = ur[r1+0]*hi0 + ui[r1+0]*hr0 + ur[r1+1]*hi1 + ui[r1+1]*hr1
              + ur[r1+2]*hi2 + ui[r1+2]*hr2 + ur[r1+3]*hi3 + ui[r1+3]*hr3;

    v2f b_re, b_im;
    b_re.x = rw*xr0; b_re.y = rw*xr1;
    b_im.x = rw*xi0; b_im.y = rw*xi1;

    accR  = wmma_f32x4(a_re, b_re, accR);    // Hr^T Gr
    accR  = wmma_f32x4(a_im, b_im, accR);    // + Hi^T Gi
    accIp = wmma_f32x4(a_re, b_im, accIp);   // Hr^T Gi
    accIm = wmma_f32x4(a_im, b_re, accIm);   // Hi^T Gr (subtract later)
  }
  int rbase = trow*16 + ((l >> 4) << 3);
  float* outR = ULre + (size_t)bc*MAT;
  float* outI = ULim + (size_t)bc*MAT;
#pragma unroll
  for (int r = 0; r < 8; ++r) {
    outR[(rbase + r)*MDIM + col] = accR[r];
    outI[(rbase + r)*MDIM + col] = accIp[r] - accIm[r];
  }
}

// ---------------------------------------------------------------------------
// K5: parallel cyclic Jacobi eigensolver, 128x128 Hermitian, all in LDS.
// Overwrites UL buffer with eigenvectors E[m][i]; lam = max(diag, 0).
// ---------------------------------------------------------------------------
extern __shared__ float eig_smem[];

__global__ void k_eig(float* __restrict__ ULre, float* __restrict__ ULim,
                      float* __restrict__ lam) {
  int bc = blockIdx.x;
  int tid = threadIdx.x;                  // 256 threads
  float* Ar = eig_smem;
  float* Ai = eig_smem + 1*MDIM*EP;
  float* Er = eig_smem + 2*MDIM*EP;
  float* Ei = eig_smem + 3*MDIM*EP;
  __shared__ float cs[64], srot[64], sirot[64];
  __shared__ int pp[64], qq[64];

  float* gre = ULre + (size_t)bc*MAT;
  float* gim = ULim + (size_t)bc*MAT;

  for (int idx = tid; idx < MAT; idx += 256) {   // load + hermitianize
    int m = idx >> 7, i = idx & 127;
    Ar[m*EP+i] = 0.5f*(gre[m*MDIM+i] + gre[i*MDIM+m]);
    Ai[m*EP+i] = 0.5f*(gim[m*MDIM+i] - gim[i*MDIM+m]);
    Er[m*EP+i] = (m == i) ? 1.f : 0.f;
    Ei[m*EP+i] = 0.f;
  }
  __syncthreads();

  for (int sweep = 0; sweep < NSWEEP; ++sweep) {
    for (int r = 0; r < 127; ++r) {
      if (tid < 64) {
        int p, q;
        if (tid == 0) { p = 127; q = r; }
        else { p = (r + tid) % 127; q = (r - tid + 127) % 127; }
        pp[tid] = p; qq[tid] = q;
        float app = Ar[p*EP+p], aqq = Ar[q*EP+q];
        float brE = Ar[p*EP+q], biE = Ai[p*EP+q];
        float b2 = brE*brE + biE*biE;
        float cc = 1.f, sr_ = 0.f, si_ = 0.f;
        if (b2 > 1e-36f) {
          float bm = sqrtf(b2);
          float tau = (aqq - app) / (2.f*bm);
          float tt = ((tau >= 0.f) ? 1.f : -1.f) / (fabsf(tau) + sqrtf(1.f + tau*tau));
          cc = 1.f / sqrtf(1.f + tt*tt);
          float sg = tt*cc;
          sr_ = sg*brE/bm; si_ = sg*biE/bm;
        }
        cs[tid] = cc; srot[tid] = sr_; sirot[tid] = si_;
      }
      __syncthreads();
      // column update: A <- A J, E <- E J
      for (int w = tid; w < 64*128; w += 256) {
        int j = w >> 7, row = w & 127;
        int p = pp[j], q = qq[j];
        float cc = cs[j], sr_ = srot[j], si_ = sirot[j];
        float xr = Ar[row*EP+p], xi = Ai[row*EP+p];
        float yr = Ar[row*EP+q], yi = Ai[row*EP+q];
        Ar[row*EP+p] = cc*xr - (sr_*yr + si_*yi);          // c*x - conj(s)*y
        Ai[row*EP+p] = cc*xi - (sr_*yi - si_*yr);
        Ar[row*EP+q] = sr_*xr - si_*xi + cc*yr;            // s*x + c*y
        Ai[row*EP+q] = sr_*xi + si_*xr + cc*yi;
        xr = Er[row*EP+p]; xi = Ei[row*EP+p];
        yr = Er[row*EP+q]; yi = Ei[row*EP+q];
        Er[row*EP+p] = cc*xr - (sr_*yr + si_*yi);
        Ei[row*EP+p] = cc*xi - (sr_*yi - si_*yr);
        Er[row*EP+q] = sr_*xr - si_*xi + cc*yr;
        Ei[row*EP+q] = sr_*xi + si_*xr + cc*yi;
      }
      __syncthreads();
      // row update: A <- J^H A
      for (int w = tid; w < 64*128; w += 256) {
        int j = w >> 7, colx = w & 127;
        int p = pp[j], q = qq[j];
        float cc = cs[j], sr_ = srot[j], si_ = sirot[j];
        float xr = Ar[p*EP+colx], xi = Ai[p*EP+colx];
        float yr = Ar[q*EP+colx], yi = Ai[q*EP+colx];
        Ar[p*EP+colx] = cc*xr - (sr_*yr - si_*yi);         // c*x - s*y
        Ai[p*EP+colx] = cc*xi - (sr_*yi + si_*yr);
        Ar[q*EP+colx] = sr_*xr + si_*xi + cc*yr;           // conj(s)*x + c*y
        Ai[q*EP+colx] = sr_*xi - si_*xr + cc*yi;
      }
      __syncthreads();
    }
  }
  // write back: E overwrites UL; lam = clamp(diag, 0)
  for (int idx = tid; idx < MAT; idx += 256) {
    int m = idx >> 7, i = idx & 127;
    gre[m*MDIM+i] = Er[m*EP+i];
    gim[m*MDIM+i] = Ei[m*EP+i];
  }
  if (tid < 128) lam[bc*MDIM + tid] = fmaxf(Ar[tid*EP+tid], 0.f);
}

// ---------------------------------------------------------------------------
// K6 (WMMA + async LDS staging): T = S[b] * E[b,c];
// c[b,c][i] += masked sum_q |T[q][i]|^2.  A- and B-strips staged in LDS.
// ---------------------------------------------------------------------------
__global__ void k_cvec(const float* __restrict__ Sre, const float* __restrict__ Sim,
                       const float* __restrict__ Ere, const float* __restrict__ Eim,
                       const int* __restrict__ asg, float* __restrict__ cbuf) {
  int t = blockIdx.x;
  int tile = t & 63;
  int bc = t >> 6;
  int b = bc >> 3, c = bc & 7;
  int trow = tile >> 3, tcol = tile & 7;       // trow: q-block, tcol: i-block
  int l  = threadIdx.x;
  int lr = l & 15;
  int kb = (l >> 4) << 1;
  int col = tcol*16 + lr;
  const float* SreB = Sre + (size_t)b*MAT;
  const float* SimB = Sim + (size_t)b*MAT;
  const float* EreB = Ere + (size_t)bc*MAT;
  const float* EimB = Eim + (size_t)bc*MAT;

  __shared__ float sAre[16*AP], sAim[16*AP];   // S-strip rows  [16][128]
  __shared__ float sBre[128*16], sBim[128*16]; // E-strip cols  [128][16]
  {
    const float* gre = SreB + (trow*16)*MDIM;  // contiguous 8KB per plane
    const float* gim = SimB + (trow*16)*MDIM;
    unsigned int oARe = lds_off_of(sAre);
    unsigned int oAIm = lds_off_of(sAim);
    for (int r = 0; r < 16; ++r) {
      async_copy16(oARe + (unsigned)(r*AP + l*4)*4u, gre + r*MDIM + l*4);
      async_copy16(oAIm + (unsigned)(r*AP + l*4)*4u, gim + r*MDIM + l*4);
    }
    // B-strip: 128 rows x 16 cols; 4 lanes per row -> 8 rows per issue
    unsigned int oBRe = lds_off_of(sBre);
    unsigned int oBIm = lds_off_of(sBim);
    int brow = l >> 2;
    int bch  = (l & 3) * 4;
    for (int it = 0; it < 16; ++it) {
      int row = it*8 + brow;
      async_copy16(oBRe + (unsigned)(row*16 + bch)*4u,
                   EreB + row*MDIM + tcol*16 + bch);
      async_copy16(oBIm + (unsigned)(row*16 + bch)*4u,
                   EimB + row*MDIM + tcol*16 + bch);
    }
  }
  wait_async0();

  v8f acc1 = {}, acc2 = {}, acc3 = {};
  for (int kk = 0; kk < 32; ++kk) {
    int m0 = kk*4;
    v2f a_re = *(const v2f*)(sAre + lr*AP + m0 + kb);
    v2f a_im = *(const v2f*)(sAim + lr*AP + m0 + kb);
    v2f b_re, b_im;
    b_re.x = sBre[(m0+kb  )*16 + lr];
    b_re.y = sBre[(m0+kb+1)*16 + lr];
    b_im.x = sBim[(m0+kb  )*16 + lr];
    b_im.y = sBim[(m0+kb+1)*16 + lr];
    acc1 = wmma_f32x4(a_re, b_re, acc1);   // Sr Er
    acc2 = wmma_f32x4(a_im, b_im, acc2);   // Si Ei (subtract)
    acc3 = wmma_f32x4(a_re, b_im, acc3);   // Sr Ei
    acc3 = wmma_f32x4(a_im, b_re, acc3);   // + Si Er
  }
  float part = 0.f;
  int rbase = trow*16 + ((l >> 4) << 3);
#pragma unroll
  for (int r = 0; r < 8; ++r) {
    int qg = rbase + r;
    if (asg[qg >> 2] == c) {
      float tr = acc1[r] - acc2[r];
      float ti = acc3[r];
      part += tr*tr + ti*ti;
    }
  }
  atomicAdd(&cbuf[bc*MDIM + col], part);
}

// ---------------------------------------------------------------------------
// K7: bisection for mu, then laminv = 1/(lam + mu) with pseudo-inverse guard
// ---------------------------------------------------------------------------
__global__ void k_mu(const float* __restrict__ lam, const float* __restrict__ cbuf,
                     const float* __restrict__ bss_pow, float* __restrict__ laminv) {
  int bc = blockIdx.x;
  int b = bc >> 3, c = bc & 7;
  int i = threadIdx.x;                   // 128
  __shared__ float red[128];
  float li = lam[bc*MDIM + i];
  float ci = fmaxf(cbuf[bc*MDIM + i], 0.f) / bss_pow[b*NBSS + c];

  red[i] = ci; __syncthreads();
  for (int s = 64; s > 0; s >>= 1) { if (i < s) red[i] += red[i+s]; __syncthreads(); }
  float hi = sqrtf(red[0]);
  float lo = 0.f;
  __syncthreads();
  for (int it = 0; it < 8; ++it) {
    float mid = 0.5f*(lo + hi);
    float d = li + mid + 1e-20f;
    red[i] = ci / (d*d);
    __syncthreads();
    for (int s = 64; s > 0; s >>= 1) { if (i < s) red[i] += red[i+s]; __syncthreads(); }
    float f = red[0];
    __syncthreads();
    bool gt = f > 1.f;
    lo = gt ? mid : lo;
    hi = gt ? hi : mid;
  }
  float mu = fmaxf(0.5f*(lo + hi), 0.f);
  float ll = li + mu;
  if (ll < 1e-12f) ll = 1e12f;
  laminv[bc*MDIM + i] = 1.f / ll;
}

// ---------------------------------------------------------------------------
// K8: v_out[b,k] = E diag(laminv) E^H vt  (complex64 interleaved output)
// ---------------------------------------------------------------------------
__global__ void k_out(const float* __restrict__ Sre, const float* __restrict__ Sim,
                      const float* __restrict__ Ere, const float* __restrict__ Eim,
                      const float* __restrict__ laminv, const int* __restrict__ asg,
                      float2* __restrict__ out) {
  int bk = blockIdx.x;
  int b = bk >> 5, k = bk & 31;
  int tid = threadIdx.x;                 // 128
  int c0 = asg[k];
  __shared__ float vtr[4*128], vti[4*128], ysr[4*128], ysi[4*128];
  for (int p = 0; p < 4; ++p) {
    vtr[p*128+tid] =  Sre[(size_t)b*MAT + (k*4+p)*MDIM + tid];
    vti[p*128+tid] = -Sim[(size_t)b*MAT + (k*4+p)*MDIM + tid];
  }
  __syncthreads();
  const float* EreP = Ere + (size_t)(b*NBSS + c0)*MAT;
  const float* EimP = Eim + (size_t)(b*NBSS + c0)*MAT;
  // phase 1: y[i] = conj(E[:,i]) . vt   (tid == i)
  float yr[4] = {0,0,0,0}, yi[4] = {0,0,0,0};
  for (int m = 0; m < MDIM; ++m) {
    float er = EreP[m*MDIM + tid], ei = EimP[m*MDIM + tid];
    for (int p = 0; p < 4; ++p) {
      float vr = vtr[p*128+m], vi = vti[p*128+m];
      yr[p] += er*vr + ei*vi;
      yi[p] += er*vi - ei*vr;
    }
  }
  float li = laminv[(b*NBSS + c0)*MDIM + tid];
  for (int p = 0; p < 4; ++p) { ysr[p*128+tid] = yr[p]*li; ysi[p*128+tid] = yi[p]*li; }
  __syncthreads();
  // phase 2: out[m] = E[m,:] . y   (tid == m)
  float outr[4] = {0,0,0,0}, outi[4] = {0,0,0,0};
  for (int i2 = 0; i2 < MDIM; ++i2) {
    float er = EreP[tid*MDIM + i2], ei = EimP[tid*MDIM + i2];
    for (int p = 0; p < 4; ++p) {
      float sr = ysr[p*128+i2], si = ysi[p*128+i2];
      outr[p] += er*sr - ei*si;
      outi[p] += er*si + ei*sr;
    }
  }
  for (int p = 0; p < 4; ++p)
    out[((size_t)bk*MDIM + tid)*NDIM + p] = float2{outr[p], outi[p]};
}

// ---------------------------------------------------------------------------
// Launch
// ---------------------------------------------------------------------------
extern "C" void kernel_launch(void* const* d_in, const int* in_sizes, int n_in,
                              void* d_out, int out_size, void* d_ws, size_t ws_size,
                              hipStream_t stream) {
  const float* v_re   = (const float*)d_in[0];
  const float* v_im   = (const float*)d_in[1];
  const float* H_re   = (const float*)d_in[2];
  const float* H_im   = (const float*)d_in[3];
  const float* noise  = (const float*)d_in[4];
  const float* rw     = (const float*)d_in[5];
  const float* bsspow = (const float*)d_in[6];
  const int*   asg    = (const int*)d_in[7];

  float* ws = (float*)d_ws;
  size_t o = 0;
  float* covr = ws + o; o += BK*16;
  float* covi = ws + o; o += BK*16;
  float* hdvr = ws + o; o += BK*16;
  float* hdvi = ws + o; o += BK*16;
  float* uwr  = ws + o; o += BK*16;
  float* uwi  = ws + o; o += BK*16;
  float* uwur = ws + o; o += BK*16;
  float* uwui = ws + o; o += BK*16;
  float* Sre  = ws + o; o += (size_t)BATCH*MAT;
  float* Sim  = ws + o; o += (size_t)BATCH*MAT;
  float* HTre = ws + o; o += (size_t)NBSS*MAT;
  float* HTim = ws + o; o += (size_t)NBSS*MAT;
  float* ULre = ws + o; o += (size_t)BC*MAT;    // becomes E after k_eig
  float* ULim = ws + o; o += (size_t)BC*MAT;
  float* lamb = ws + o; o += BC*MDIM;
  float* lmin = ws + o; o += BC*MDIM;
  float* cbuf = ws + o; o += BC*MDIM;

  (void)hipFuncSetAttribute(reinterpret_cast<const void*>(k_eig),
                            hipFuncAttributeMaxDynamicSharedMemorySize,
                            EIG_LDS_BYTES);

  k_transposeH<<<NBSS*MDIM, MDIM, 0, stream>>>(H_re, H_im, HTre, HTim);
  k_cov<<<BK, 128, 0, stream>>>(v_re, v_im, H_re, H_im, noise, asg,
                                covr, covi, hdvr, hdvi);
  k_small<<<(BK+255)/256, 256, 0, stream>>>(covr, covi, hdvr, hdvi,
                                            uwr, uwi, uwur, uwui);
  k_vt<<<BK, MDIM, 0, stream>>>(H_re, H_im, uwr, uwi, rw, asg, Sre, Sim);
  k_ul<<<BC*64, 32, 0, stream>>>(HTre, HTim, H_re, H_im, uwur, uwui, rw,
                                 ULre, ULim);
  k_eig<<<BC, 256, EIG_LDS_BYTES, stream>>>(ULre, ULim, lamb);
  (void)hipMemsetAsync(cbuf, 0, (size_t)BC*MDIM*sizeof(float), stream);
  k_cvec<<<BC*64, 32, 0, stream>>>(Sre, Sim, ULre, ULim, asg, cbuf);
  k_mu<<<BC, MDIM, 0, stream>>>(lamb, cbuf, bsspow, lmin);
  k_out<<<BK, MDIM, 0, stream>>>(Sre, Sim, ULre, ULim, lmin, asg,
                                 (float2*)d_out);
}